// MultiHeadTemporalAttention_58866821759676
// MI455X (gfx1250) — compile-verified
//
#include <hip/hip_runtime.h>

#define NN 100000   // nodes
#define NE 600000   // edges
#define HID 128
#define NH 8
#define HD 16
#define NT 32
#define RT 8        // row tiles (of 16) per block in GEMM kernels

typedef float v2f __attribute__((ext_vector_type(2)));
typedef float v8f __attribute__((ext_vector_type(8)));

__device__ __forceinline__ v8f wmma_f32_4(v2f a, v2f b, v8f c) {
    // V_WMMA_F32_16X16X4_F32 : D(16x16 f32) = A(16x4 f32) * B(4x16 f32) + C
    return __builtin_amdgcn_wmma_f32_16x16x4_f32(
        /*neg_a=*/false, a, /*neg_b=*/false, b,
        /*c_mod=*/(short)0, c, /*reuse_a=*/false, /*reuse_b=*/false);
}

// ---------------------------------------------------------------------------
// init: zero agg + segsum, segmax = INT_MIN (signed monotonic -inf)
// ---------------------------------------------------------------------------
__global__ __launch_bounds__(256) void tga_init_kernel(float* __restrict__ agg,
                                                       float* __restrict__ segsum,
                                                       int* __restrict__ segmax) {
    int t = blockIdx.x * blockDim.x + threadIdx.x;
    if (t < NN * HID) agg[t] = 0.0f;
    if (t < NN * NH) { segsum[t] = 0.0f; segmax[t] = (int)0x80000000; }
}

// ---------------------------------------------------------------------------
// Generic [rows,128] @ [128,128] + bias with register-cached weight fragments.
// One wave owns a 16-col stripe; 8 waves/block cover 128 cols; RT=8 row tiles
// per block reuse the cached B fragments (one global_load_b64 per WMMA).
// ---------------------------------------------------------------------------
__device__ __forceinline__ void gemm128_block(
    const float* __restrict__ X, const float* __restrict__ W,
    const float* __restrict__ B, float* __restrict__ O, int rows) {
    const int wave = threadIdx.x >> 5;
    const int lane = threadIdx.x & 31;
    const int half = lane >> 4;          // K-pair select (ISA 16x4 f32 A/B layout)
    const int r    = lane & 15;          // M for A, N for B/C/D
    const int col0 = wave * 16;

    // cache all 32 B fragments (K=128 in steps of 4) in VGPRs
    v2f bf[32];
#pragma unroll
    for (int ks = 0; ks < 32; ++ks) {
        const int ka = ks * 4 + 2 * half;
        bf[ks] = v2f{ W[ka * HID + col0 + r], W[(ka + 1) * HID + col0 + r] };
    }
    const float bias = B[col0 + r];

    for (int t = 0; t < RT; ++t) {
        const int row0 = (blockIdx.x * RT + t) * 16;
        if (row0 >= rows) break;
        const float* xrow = X + (size_t)(row0 + r) * HID + 2 * half;
        v8f acc = {};
#pragma unroll
        for (int ks = 0; ks < 32; ++ks) {
            v2f a = v2f{ xrow[ks * 4], xrow[ks * 4 + 1] };   // A[M=r][4ks+2h .. +1]
            acc = wmma_f32_4(a, bf[ks], acc);
        }
#pragma unroll
        for (int v = 0; v < 8; ++v)      // D element (M = v + 8*half, N = r)
            O[(size_t)(row0 + v + 8 * half) * HID + col0 + r] = acc[v] + bias;
    }
}

__global__ __launch_bounds__(256) void tga_qkv_proj_kernel(
    const float* __restrict__ q_in, const float* __restrict__ k_in, const float* __restrict__ v_in,
    const float* __restrict__ Wq, const float* __restrict__ bq,
    const float* __restrict__ Wk, const float* __restrict__ bk,
    const float* __restrict__ Wv, const float* __restrict__ bv,
    float* __restrict__ Qo, float* __restrict__ Ko, float* __restrict__ Vo) {
    if (blockIdx.y == 0)      gemm128_block(q_in, Wq, bq, Qo, NN);
    else if (blockIdx.y == 1) gemm128_block(k_in, Wk, bk, Ko, NN);
    else                      gemm128_block(v_in, Wv, bv, Vo, NN);
}

__global__ __launch_bounds__(256) void tga_out_proj_kernel(
    const float* __restrict__ A, const float* __restrict__ W,
    const float* __restrict__ B, float* __restrict__ O) {
    gemm128_block(A, W, B, O, NN);
}

// ---------------------------------------------------------------------------
// Fused temporal projection + per-head dot:
// tq = TE@Wtq+btq, tk = TE@Wtk+btk (registers only), score[e,h] = sum_d tq*tk.
// Wave == head (16-col stripe). Weight fragments cached; RT row tiles/block.
// ---------------------------------------------------------------------------
__global__ __launch_bounds__(256) void tga_temporal_score_kernel(
    const float* __restrict__ TE,
    const float* __restrict__ Wtq, const float* __restrict__ btq,
    const float* __restrict__ Wtk, const float* __restrict__ btk,
    float* __restrict__ scores) {
    const int h    = threadIdx.x >> 5;   // head == wave
    const int lane = threadIdx.x & 31;
    const int half = lane >> 4;
    const int r    = lane & 15;
    const int col0 = h * HD;

    v2f bfq[8], bfk[8];
#pragma unroll
    for (int ks = 0; ks < 8; ++ks) {
        const int ka = ks * 4 + 2 * half;
        bfq[ks] = v2f{ Wtq[ka * HID + col0 + r], Wtq[(ka + 1) * HID + col0 + r] };
        bfk[ks] = v2f{ Wtk[ka * HID + col0 + r], Wtk[(ka + 1) * HID + col0 + r] };
    }
    const float biasq = btq[col0 + r];
    const float biask = btk[col0 + r];

    for (int t = 0; t < RT; ++t) {
        const int row0 = (blockIdx.x * RT + t) * 16;
        if (row0 >= NE) break;
        const float* trow = TE + (size_t)(row0 + r) * NT + 2 * half;
        v8f cq = {}, ck = {};
#pragma unroll
        for (int ks = 0; ks < 8; ++ks) {
            v2f a = v2f{ trow[ks * 4], trow[ks * 4 + 1] };
            cq = wmma_f32_4(a, bfq[ks], cq);
            ck = wmma_f32_4(a, bfk[ks], ck);
        }
#pragma unroll
        for (int v = 0; v < 8; ++v) {
            float p = (cq[v] + biasq) * (ck[v] + biask);  // element (M=v+8*half, N=r)
            p += __shfl_xor(p, 1, 32);                    // reduce over N (16 lanes/half)
            p += __shfl_xor(p, 2, 32);
            p += __shfl_xor(p, 4, 32);
            p += __shfl_xor(p, 8, 32);
            if (r == 0)
                scores[(size_t)(row0 + v + 8 * half) * NH + h] = p;
        }
    }
}

// ---------------------------------------------------------------------------
// Per-(edge,head) gathered Q.K dot (L2-resident: Q+K = 102MB < 192MB L2)
// ---------------------------------------------------------------------------
__global__ __launch_bounds__(256) void tga_qk_score_kernel(
    const float* __restrict__ Q, const float* __restrict__ K,
    const int* __restrict__ ei, float* __restrict__ scores) {
    int t = blockIdx.x * blockDim.x + threadIdx.x;
    if (t >= NE * NH) return;
    const int e = t >> 3, h = t & 7;
    const int src = ei[e];
    const int tgt = ei[NE + e];
    const float4* qp = (const float4*)(Q + (size_t)src * HID + h * HD);
    const float4* kp = (const float4*)(K + (size_t)tgt * HID + h * HD);
    float s = 0.0f;
#pragma unroll
    for (int i = 0; i < 4; ++i) {
        float4 a = qp[i], b = kp[i];
        s += a.x * b.x + a.y * b.y + a.z * b.z + a.w * b.w;
    }
    scores[t] += s * 0.25f;   // 1/sqrt(HD=16)
}

__device__ __forceinline__ int mono_map(float x) {
    int u = __float_as_int(x);
    return (u >= 0) ? u : (u ^ 0x7FFFFFFF);   // signed-monotonic float ordering
}
__device__ __forceinline__ float mono_unmap(int m) {
    return __int_as_float((m >= 0) ? m : (m ^ 0x7FFFFFFF));
}

__global__ __launch_bounds__(256) void tga_segmax_kernel(
    const float* __restrict__ scores, const int* __restrict__ ei, int* __restrict__ segmax) {
    int t = blockIdx.x * blockDim.x + threadIdx.x;
    if (t >= NE * NH) return;
    const int e = t >> 3, h = t & 7;
    const int tgt = ei[NE + e];
    atomicMax(&segmax[tgt * NH + h], mono_map(scores[t]));
}

__global__ __launch_bounds__(256) void tga_expsum_kernel(
    float* __restrict__ scores, const int* __restrict__ ei,
    const int* __restrict__ segmax, float* __restrict__ segsum) {
    int t = blockIdx.x * blockDim.x + threadIdx.x;
    if (t >= NE * NH) return;
    const int e = t >> 3, h = t & 7;
    const int tgt = ei[NE + e];
    const float mx = mono_unmap(segmax[tgt * NH + h]);
    const float ex = __expf(scores[t] - mx);
    scores[t] = ex;
    atomicAdd(&segsum[tgt * NH + h], ex);
}

// normalize -> attn_weights output, then scatter-add w * V[src] into agg[tgt]
__global__ __launch_bounds__(256) void tga_aggregate_kernel(
    const float* __restrict__ scores, const int* __restrict__ ei,
    const float* __restrict__ segsum, const float* __restrict__ V,
    float* __restrict__ agg, float* __restrict__ attn_w) {
    int t = blockIdx.x * blockDim.x + threadIdx.x;
    if (t >= NE * NH) return;
    const int e = t >> 3, h = t & 7;
    const int src = ei[e];
    const int tgt = ei[NE + e];
    const float w = scores[t] / segsum[tgt * NH + h];
    attn_w[t] = w;
    const float* vp = V + (size_t)src * HID + h * HD;
    float* ap = agg + (size_t)tgt * HID + h * HD;
#pragma unroll
    for (int d = 0; d < HD; ++d)
        atomicAdd(&ap[d], w * vp[d]);
}

extern "C" void kernel_launch(void* const* d_in, const int* in_sizes, int n_in,
                              void* d_out, int out_size, void* d_ws, size_t ws_size,
                              hipStream_t stream) {
    const float* query = (const float*)d_in[0];
    const float* key   = (const float*)d_in[1];
    const float* value = (const float*)d_in[2];
    const float* te    = (const float*)d_in[3];
    const int*   ei    = (const int*)d_in[4];
    const float* Wq  = (const float*)d_in[5];  const float* bq  = (const float*)d_in[6];
    const float* Wk  = (const float*)d_in[7];  const float* bk  = (const float*)d_in[8];
    const float* Wv  = (const float*)d_in[9];  const float* bv  = (const float*)d_in[10];
    const float* Wtq = (const float*)d_in[11]; const float* btq = (const float*)d_in[12];
    const float* Wtk = (const float*)d_in[13]; const float* btk = (const float*)d_in[14];
    const float* Wout = (const float*)d_in[15]; const float* bout = (const float*)d_in[16];

    float* out      = (float*)d_out;
    float* attn_out = out;                        // [NN,128]
    float* attn_w   = out + (size_t)NN * HID;     // [NE,8]

    float* ws     = (float*)d_ws;
    float* Q      = ws;
    float* K      = Q + (size_t)NN * HID;
    float* V      = K + (size_t)NN * HID;
    float* agg    = V + (size_t)NN * HID;
    float* scores = agg + (size_t)NN * HID;       // [NE,8]
    float* segsum = scores + (size_t)NE * NH;     // [NN,8]
    int*   segmax = (int*)(segsum + (size_t)NN * NH);

    const dim3 blk(256);
    const int eh_blocks   = (NE * NH + 255) / 256;
    const int node_tiles  = NN / 16;                       // 6250
    const int edge_tiles  = NE / 16;                       // 37500
    const int node_blocks = (node_tiles + RT - 1) / RT;    // 782
    const int edge_blocks = (edge_tiles + RT - 1) / RT;    // 4688

    tga_init_kernel<<<(NN * HID + 255) / 256, blk, 0, stream>>>(agg, segsum, segmax);
    tga_qkv_proj_kernel<<<dim3(node_blocks, 3), blk, 0, stream>>>(
        query, key, value, Wq, bq, Wk, bk, Wv, bv, Q, K, V);
    tga_temporal_score_kernel<<<edge_blocks, blk, 0, stream>>>(te, Wtq, btq, Wtk, btk, scores);
    tga_qk_score_kernel<<<eh_blocks, blk, 0, stream>>>(Q, K, ei, scores);
    tga_segmax_kernel<<<eh_blocks, blk, 0, stream>>>(scores, ei, segmax);
    tga_expsum_kernel<<<eh_blocks, blk, 0, stream>>>(scores, ei, segmax, segsum);
    tga_aggregate_kernel<<<eh_blocks, blk, 0, stream>>>(scores, ei, segsum, V, agg, attn_w);
    tga_out_proj_kernel<<<node_blocks, blk, 0, stream>>>(agg, Wout, bout, attn_out);
}